// QGate_65481071410789
// MI455X (gfx1250) — compile-verified
//
#include <hip/hip_runtime.h>
#include <cstdint>

typedef __attribute__((ext_vector_type(2))) float v2f;
typedef __attribute__((ext_vector_type(8))) float v8f;

#define BROWS    131072
#define FDIM     512
#define NQ       20
#define ROWS_BLK 128                  // rows of x per workgroup (8 waves x 16)
#define KCHUNK   64                   // K columns staged per LDS chunk
#define NCHUNK   (FDIM / KCHUNK)      // 8
#define WPAD     516                  // W row stride in LDS (bank-conflict pad)
#define XPAD     68                   // x row stride in LDS (bank-conflict pad)
#define WS_ELEMS (NQ * WPAD)          // 10320 floats
#define XSBUF    (ROWS_BLK * XPAD)    // 8704 floats per buffer
#define SMEM_ELEMS (WS_ELEMS + 2 * XSBUF)  // 27728 floats = 110912 B

__device__ __forceinline__ uint32_t lds_off(const void* p) {
  // Generic LDS pointer = {shared aperture, lds_offset[31:0]}; truncation yields LDS byte offset.
  return (uint32_t)(uintptr_t)p;
}

__device__ __forceinline__ void async_b128(uint32_t lds, const float* g) {
  asm volatile("global_load_async_to_lds_b128 %0, %1, off"
               :: "v"(lds), "v"((unsigned long long)(uintptr_t)g)
               : "memory");
}

// Stage a 128-row x KCHUNK-col fp32 tile of x into LDS with async b128 copies.
// 256 threads x 8 instructions, fully coalesced 16B/lane.
__device__ __forceinline__ void issue_chunk(const float* __restrict__ x,
                                            int row0, int f0,
                                            uint32_t xbuf_lds, int tid) {
#pragma unroll
  for (int p = 0; p < 8; ++p) {
    int id  = (p << 8) + tid;       // 0..2047
    int row = id >> 4;              // 0..127
    int c4  = (id & 15) << 2;       // float col 0,4,...,60
    const float* g = x + (size_t)(row0 + row) * FDIM + f0 + c4;
    async_b128(xbuf_lds + (uint32_t)((row * XPAD + c4) << 2), g);
  }
}

__global__ __launch_bounds__(256) void qgate_cumprod_kernel(
    const float* __restrict__ x, const float* __restrict__ W,
    float* __restrict__ out) {
  __shared__ __align__(16) float smem[SMEM_ELEMS];
  const int tid  = threadIdx.x;
  const int row0 = blockIdx.x * ROWS_BLK;

  const uint32_t xls0 = lds_off(&smem[WS_ELEMS]);
  const uint32_t xls1 = lds_off(&smem[WS_ELEMS + XSBUF]);

  // Kick off async copy of chunk 0 immediately, overlap with W staging.
  issue_chunk(x, row0, 0, xls0, tid);

  // Stage W rows 0..19 (hits L2 after first block); padded row stride.
  for (int i = tid; i < NQ * FDIM; i += 256) {
    int q = i >> 9;                 // /512
    int f = i & (FDIM - 1);
    smem[q * WPAD + f] = W[i];
  }

  const int lane = tid & 31;        // wave32
  const int wid  = tid >> 5;        // 0..7
  const int lr   = lane & 15;
  const int lh   = lane >> 4;       // lane half

  v8f acc0 = {};                    // D[q=0..15][b tile]
  v8f acc1 = {};                    // D[q=16..31][b tile] (q>=20 unused)

  const int arow0 = lr * WPAD;            // A tile0: W row q=lr
  const int arow1 = (16 + lr) * WPAD;     // A tile1: rows>=20 read stale LDS; only feeds unused D rows
  const int brow  = (wid * 16 + lr) * XPAD;

  for (int c = 0; c < NCHUNK; ++c) {
    if (c + 1 < NCHUNK) {
      issue_chunk(x, row0, (c + 1) * KCHUNK, ((c + 1) & 1) ? xls1 : xls0, tid);
      asm volatile("s_wait_asynccnt 0x8" ::: "memory");  // chunk c landed, c+1 in flight
    } else {
      asm volatile("s_wait_asynccnt 0x0" ::: "memory");
    }
    __syncthreads();

    const float* Xb   = &smem[WS_ELEMS + ((c & 1) ? XSBUF : 0)];
    const int   fbase = c * KCHUNK;
#pragma unroll
    for (int s = 0; s < KCHUNK / 4; ++s) {
      const int f  = fbase + 4 * s + 2 * lh;   // absolute K for A frag
      const int cl = 4 * s + 2 * lh;           // chunk-local K for B frag
      v2f bfrag = *(const v2f*)&Xb[brow + cl];        // x^T 4x16 fragment
      v2f a0    = *(const v2f*)&smem[arow0 + f];      // W 16x4 fragment (q 0..15)
      v2f a1    = *(const v2f*)&smem[arow1 + f];      // W 16x4 fragment (q 16..31)
      acc0 = __builtin_amdgcn_wmma_f32_16x16x4_f32(false, a0, false, bfrag,
                                                   (short)0, acc0, false, false);
      acc1 = __builtin_amdgcn_wmma_f32_16x16x4_f32(false, a1, false, bfrag,
                                                   (short)0, acc1, false, false);
    }
    __syncthreads();                 // buffer (c&1) safe to overwrite at c+2
  }

  // Epilogue: cos + cumprod along q.
  // Lane lh=0 holds q=0..7 (acc0) and q=16..23 (acc1); lane lh=1 holds q=8..15.
  float pre[8];
  float run = 1.0f;
#pragma unroll
  for (int v = 0; v < 8; ++v) { run *= __cosf(acc0[v]); pre[v] = run; }
  const float partner = __shfl_xor(run, 16, 32);  // other half's 8-term product

  const int b = row0 + wid * 16 + lr;             // D column N -> batch row
  float* o = out + (size_t)b * NQ;
  if (lh == 0) {
    *(float4*)(o + 0) = make_float4(pre[0], pre[1], pre[2], pre[3]);
    *(float4*)(o + 4) = make_float4(pre[4], pre[5], pre[6], pre[7]);
    float r = run * partner;                      // prod cos(q=0..15)
    float t[4];
#pragma unroll
    for (int v = 0; v < 4; ++v) { r *= __cosf(acc1[v]); t[v] = r; }
    *(float4*)(o + 16) = make_float4(t[0], t[1], t[2], t[3]);
  } else {
    float t[8];
#pragma unroll
    for (int v = 0; v < 8; ++v) t[v] = partner * pre[v];  // prefix(q<8) * local cumprod
    *(float4*)(o + 8)  = make_float4(t[0], t[1], t[2], t[3]);
    *(float4*)(o + 12) = make_float4(t[4], t[5], t[6], t[7]);
  }
}

extern "C" void kernel_launch(void* const* d_in, const int* in_sizes, int n_in,
                              void* d_out, int out_size, void* d_ws, size_t ws_size,
                              hipStream_t stream) {
  (void)in_sizes; (void)n_in; (void)d_ws; (void)ws_size; (void)out_size;
  const float* x = (const float*)d_in[0];
  const float* W = (const float*)d_in[1];
  float* out = (float*)d_out;
  dim3 grid(BROWS / ROWS_BLK);  // 1024 workgroups
  dim3 block(256);              // 8 waves (wave32)
  hipLaunchKernelGGL(qgate_cumprod_kernel, grid, block, 0, stream, x, W, out);
}